// GatNet_32091995636380
// MI455X (gfx1250) — compile-verified
//
#include <hip/hip_runtime.h>
#include <hip/hip_bf16.h>
#include <math.h>

typedef __attribute__((ext_vector_type(2))) float v2f;
typedef __attribute__((ext_vector_type(8))) float v8f;

#define D_FEAT 64
#define NEG_SLOPE 0.2f

// ---------- helpers ----------
__device__ __forceinline__ unsigned fkey(float f) {
  unsigned b = __float_as_uint(f);
  return (b & 0x80000000u) ? ~b : (b | 0x80000000u);
}
__device__ __forceinline__ float funkey(unsigned k) {
  unsigned b = (k & 0x80000000u) ? (k & 0x7FFFFFFFu) : ~k;
  return __uint_as_float(b);
}
__device__ __forceinline__ float lrelu(float x) {
  return x > 0.0f ? x : NEG_SLOPE * x;
}

// ---------- init kernels ----------
__global__ void fill_f32(float* p, float v, long n) {
  long i = (long)blockIdx.x * blockDim.x + threadIdx.x;
  if (i < n) p[i] = v;
}
__global__ void fill_u32(unsigned* p, unsigned v, long n) {
  long i = (long)blockIdx.x * blockDim.x + threadIdx.x;
  if (i < n) p[i] = v;
}

// ---------- w_s = W @ a_src, w_d = W @ a_dst (64x64 matvec, 64 threads) ----------
__global__ void matvec_kernel(const float* __restrict__ W,
                              const float* __restrict__ a_src,
                              const float* __restrict__ a_dst,
                              float* __restrict__ ws, float* __restrict__ wd) {
  int k = threadIdx.x;
  float accs = 0.0f, accd = 0.0f;
  for (int j = 0; j < D_FEAT; ++j) {
    float w = W[k * D_FEAT + j];
    accs = fmaf(w, a_src[j], accs);
    accd = fmaf(w, a_dst[j], accd);
  }
  ws[k] = accs;
  wd[k] = accd;
}

// ---------- H = X @ W via V_WMMA_F32_16X16X4_F32 ----------
// One wave computes a 16-row x 64-col tile. 8 waves / block -> 128 rows / block.
__global__ __launch_bounds__(256) void gemm_wmma_f32(const float* __restrict__ X,
                                                     const float* __restrict__ W,
                                                     float* __restrict__ H, int n_rows) {
  const int lane  = threadIdx.x & 31;
  const int wave  = threadIdx.x >> 5;
  const int mbase = (blockIdx.x * 8 + wave) * 16;
  if (mbase >= n_rows) return;              // wave-uniform: EXEC stays all-ones

  const int col   = lane & 15;              // N (and M for A-rows) within tile
  const int khalf = (lane >> 4) & 1;        // lanes 16-31 carry K offset +2

  // A: lane holds row (mbase+col), K = k0 + khalf*2 .. +1   (clamped load addr)
  int arow = mbase + col;
  if (arow >= n_rows) arow = n_rows - 1;
  const float* aptr = X + (size_t)arow * D_FEAT + khalf * 2;
  // B: lane holds column col (of N-tile), rows K = k0 + khalf*2 .. +1
  const float* bptr = W + (size_t)(khalf * 2) * D_FEAT + col;

  v8f acc0 = {}, acc1 = {}, acc2 = {}, acc3 = {};

  for (int k0 = 0; k0 < D_FEAT; k0 += 4) {
    v2f a = *(const v2f*)(aptr + k0);
    const float* bp = bptr + (size_t)k0 * D_FEAT;
    v2f b0, b1, b2, b3;
    b0.x = bp[0];  b0.y = bp[0 + D_FEAT];
    b1.x = bp[16]; b1.y = bp[16 + D_FEAT];
    b2.x = bp[32]; b2.y = bp[32 + D_FEAT];
    b3.x = bp[48]; b3.y = bp[48 + D_FEAT];
    acc0 = __builtin_amdgcn_wmma_f32_16x16x4_f32(false, a, false, b0, (short)0, acc0, false, false);
    acc1 = __builtin_amdgcn_wmma_f32_16x16x4_f32(false, a, false, b1, (short)0, acc1, false, false);
    acc2 = __builtin_amdgcn_wmma_f32_16x16x4_f32(false, a, false, b2, (short)0, acc2, false, false);
    acc3 = __builtin_amdgcn_wmma_f32_16x16x4_f32(false, a, false, b3, (short)0, acc3, false, false);
  }

  // C/D layout: VGPR r -> M = r (lanes 0-15) / r+8 (lanes 16-31), N = col
  const int moff = (lane >= 16) ? 8 : 0;
#pragma unroll
  for (int r = 0; r < 8; ++r) {
    int row = mbase + r + moff;
    if (row < n_rows) {
      float* hp = H + (size_t)row * D_FEAT + col;
      hp[0]  = acc0[r];
      hp[16] = acc1[r];
      hp[32] = acc2[r];
      hp[48] = acc3[r];
    }
  }
}

// ---------- per-node scores: s_src[i] = x[i].w_s , s_dst[i] = x[i].w_d ----------
__global__ void scores_kernel(const float* __restrict__ X,
                              const float* __restrict__ ws, const float* __restrict__ wd,
                              float* __restrict__ s_src, float* __restrict__ s_dst, int n) {
  int gw   = (int)((blockIdx.x * blockDim.x + threadIdx.x) >> 5);
  int lane = threadIdx.x & 31;
  if (gw >= n) return;
  v2f x2 = *(const v2f*)(X + (size_t)gw * D_FEAT + lane * 2);
  float ps = x2.x * ws[2 * lane] + x2.y * ws[2 * lane + 1];
  float pd = x2.x * wd[2 * lane] + x2.y * wd[2 * lane + 1];
#pragma unroll
  for (int off = 16; off; off >>= 1) {
    ps += __shfl_xor(ps, off, 32);
    pd += __shfl_xor(pd, off, 32);
  }
  if (lane == 0) { s_src[gw] = ps; s_dst[gw] = pd; }
}

// ---------- segment max over incoming edges (uint-key atomic max) ----------
__global__ void edge_max_kernel(const int* __restrict__ esrc, const int* __restrict__ edst,
                                int n_edges, int n_nodes,
                                const float* __restrict__ s_src, const float* __restrict__ s_dst,
                                unsigned* __restrict__ mkey) {
  long e = (long)blockIdx.x * blockDim.x + threadIdx.x;
  long total = (long)n_edges + n_nodes;
  if (e >= total) return;
  int s, d;
  if (e < n_edges) { s = esrc[e]; d = edst[e]; } else { s = d = (int)(e - n_edges); }
  float logit = lrelu(s_src[s] + s_dst[d]);
  atomicMax(mkey + d, fkey(logit));
}

// ---------- segment sum of exp(e - m[dst]) ----------
__global__ void edge_expsum_kernel(const int* __restrict__ esrc, const int* __restrict__ edst,
                                   int n_edges, int n_nodes,
                                   const float* __restrict__ s_src, const float* __restrict__ s_dst,
                                   const unsigned* __restrict__ mkey, float* __restrict__ denom) {
  long e = (long)blockIdx.x * blockDim.x + threadIdx.x;
  long total = (long)n_edges + n_nodes;
  if (e >= total) return;
  int s, d;
  if (e < n_edges) { s = esrc[e]; d = edst[e]; } else { s = d = (int)(e - n_edges); }
  float logit = lrelu(s_src[s] + s_dst[d]);
  float w = expf(logit - funkey(mkey[d]));
  atomicAdd(denom + d, w);
}

// ---------- out[dst] += alpha * h[src] ; one wave per edge, 2 feats/lane ----------
__global__ void edge_aggr_kernel(const int* __restrict__ esrc, const int* __restrict__ edst,
                                 int n_edges, int n_nodes,
                                 const float* __restrict__ s_src, const float* __restrict__ s_dst,
                                 const unsigned* __restrict__ mkey, const float* __restrict__ denom,
                                 const float* __restrict__ H, float* __restrict__ OUT) {
  long gw   = ((long)blockIdx.x * blockDim.x + threadIdx.x) >> 5;
  int  lane = threadIdx.x & 31;
  long total = (long)n_edges + n_nodes;
  if (gw >= total) return;
  int s, d;
  if (gw < n_edges) { s = esrc[gw]; d = edst[gw]; } else { s = d = (int)(gw - n_edges); }
  float logit = lrelu(s_src[s] + s_dst[d]);
  float alpha = expf(logit - funkey(mkey[d])) / denom[d];
  v2f hv = *(const v2f*)(H + (size_t)s * D_FEAT + lane * 2);
  float* op = OUT + (size_t)d * D_FEAT + lane * 2;
  atomicAdd(op,     alpha * hv.x);
  atomicAdd(op + 1, alpha * hv.y);
}

// ---------- one GAT layer ----------
static void run_gat_layer(const float* x, const float* W,
                          const float* a_src, const float* a_dst,
                          const int* esrc, const int* edst, int n_edges, int n_nodes,
                          float* htmp, float* s_src, float* s_dst,
                          unsigned* mkey, float* denom, float* wsv, float* wdv,
                          float* out, hipStream_t stream) {
  const long total_e = (long)n_edges + n_nodes;
  const unsigned NEG_INF_KEY = 0x007FFFFFu;  // fkey(-inf)

  long nfeat = (long)n_nodes * D_FEAT;
  fill_f32<<<dim3((unsigned)((nfeat + 255) / 256)), 256, 0, stream>>>(out, 0.0f, nfeat);
  fill_u32<<<dim3((unsigned)((n_nodes + 255) / 256)), 256, 0, stream>>>(mkey, NEG_INF_KEY, n_nodes);
  fill_f32<<<dim3((unsigned)((n_nodes + 255) / 256)), 256, 0, stream>>>(denom, 0.0f, n_nodes);

  matvec_kernel<<<1, D_FEAT, 0, stream>>>(W, a_src, a_dst, wsv, wdv);

  gemm_wmma_f32<<<dim3((unsigned)((n_nodes + 127) / 128)), 256, 0, stream>>>(x, W, htmp, n_nodes);

  scores_kernel<<<dim3((unsigned)(((long)n_nodes * 32 + 255) / 256)), 256, 0, stream>>>(
      x, wsv, wdv, s_src, s_dst, n_nodes);

  edge_max_kernel<<<dim3((unsigned)((total_e + 255) / 256)), 256, 0, stream>>>(
      esrc, edst, n_edges, n_nodes, s_src, s_dst, mkey);

  edge_expsum_kernel<<<dim3((unsigned)((total_e + 255) / 256)), 256, 0, stream>>>(
      esrc, edst, n_edges, n_nodes, s_src, s_dst, mkey, denom);

  edge_aggr_kernel<<<dim3((unsigned)((total_e * 32 + 255) / 256)), 256, 0, stream>>>(
      esrc, edst, n_edges, n_nodes, s_src, s_dst, mkey, denom, htmp, out);
}

extern "C" void kernel_launch(void* const* d_in, const int* in_sizes, int n_in,
                              void* d_out, int out_size, void* d_ws, size_t ws_size,
                              hipStream_t stream) {
  const float* x     = (const float*)d_in[0];
  const int*   edges = (const int*)d_in[1];
  const float* W1    = (const float*)d_in[2];
  const float* as1   = (const float*)d_in[3];
  const float* ad1   = (const float*)d_in[4];
  const float* W2    = (const float*)d_in[5];
  const float* as2   = (const float*)d_in[6];
  const float* ad2   = (const float*)d_in[7];
  float* out = (float*)d_out;

  const int n_nodes = in_sizes[0] / D_FEAT;   // 100000
  const int n_edges = in_sizes[1] / 2;        // 1000000
  const int* esrc = edges;
  const int* edst = edges + n_edges;

  // workspace carve-up
  float*    p     = (float*)d_ws;
  float*    h1    = p;                     p += (size_t)n_nodes * D_FEAT;
  float*    htmp  = p;                     p += (size_t)n_nodes * D_FEAT;
  float*    ssrc  = p;                     p += n_nodes;
  float*    sdst  = p;                     p += n_nodes;
  unsigned* mkey  = (unsigned*)p;          p += n_nodes;
  float*    denom = p;                     p += n_nodes;
  float*    wsv   = p;                     p += D_FEAT;
  float*    wdv   = p;                     p += D_FEAT;

  // layer 1: x -> h1
  run_gat_layer(x, W1, as1, ad1, esrc, edst, n_edges, n_nodes,
                htmp, ssrc, sdst, mkey, denom, wsv, wdv, h1, stream);
  // layer 2: h1 -> out (dropout is identity in eval mode)
  run_gat_layer(h1, W2, as2, ad2, esrc, edst, n_edges, n_nodes,
                htmp, ssrc, sdst, mkey, denom, wsv, wdv, out, stream);
}